// ResidualBlock_20126216749878
// MI455X (gfx1250) — compile-verified
//
#include <hip/hip_runtime.h>

typedef __attribute__((ext_vector_type(2))) float v2f;
typedef __attribute__((ext_vector_type(8))) float v8f;
typedef __attribute__((ext_vector_type(4))) unsigned int u32x4;
typedef __attribute__((ext_vector_type(4))) int i32x4;
typedef __attribute__((ext_vector_type(8))) int i32x8;

#define HDIM 512
#define WDIM 512
#define CCH  16
#define HT   8      // output rows per block
#define WT   64     // output cols per block
// conv1 (post-ReLU) tile + halo in LDS, channel-pair interleaved [pair][row][col][2]
#define MROWS 10
#define MCOLS 88    // 80 used; stride chosen so pair-plane = 1760 words == 32 (mod 64)
// staged input tile (conv1 input halo), TDM layout [channel][row][col]
#define IROWS 12
#define ICOLS 82

#if __has_builtin(__builtin_amdgcn_tensor_load_to_lds) && __has_builtin(__builtin_amdgcn_s_wait_tensorcnt)
#define USE_TDM 1
#else
#define USE_TDM 0
#endif

__global__ __launch_bounds__(256) void resblock_wmma_f32(
    const float* __restrict__ x,
    const float* __restrict__ w1, const float* __restrict__ b1,
    const float* __restrict__ w2, const float* __restrict__ b2,
    float* __restrict__ out)
{
    __shared__ float in_tile[CCH * IROWS * ICOLS];    // 61.5 KB: x tile + halo
    __shared__ v2f   mid2[(CCH / 2) * MROWS * MCOLS]; // 56.3 KB: relu(conv1), pair-interleaved

    const int tid  = threadIdx.x;
    const int wave = __builtin_amdgcn_readfirstlane(tid >> 5);  // uniform -> scalar loops
    const int lane = tid & 31;
    const int n    = lane & 15;   // matrix column (pixel) / A-matrix row M
    const int hi   = lane >> 4;   // lane half select

    const int blk  = blockIdx.x;
    const int wblk = blk & 7;             // WDIM/WT = 8
    const int hblk = (blk >> 3) & 63;     // HDIM/HT = 64
    const int bimg = blk >> 9;            // batch
    const int h0 = hblk * HT;
    const int w0 = wblk * WT;

    const size_t plane = (size_t)HDIM * WDIM;
    const float* xb = x   + (size_t)bimg * CCH * plane;
    float*       ob = out + (size_t)bimg * CCH * plane;

    // ============== Stage 0: stage input tile (rows h0-2..h0+9, cols w0-9..w0+72) ==============
#if USE_TDM
    const bool interior = (hblk > 0) && (hblk < 63) && (wblk > 0) && (wblk < 7);
    if (interior) {
        if (wave == 0) {
            // Tensor DMA: 3D tile 82 x 12 x 16 (x,row,channel) of f32, contiguous into LDS.
            const float* gsrc = xb + (size_t)(h0 - 2) * WDIM + (w0 - 9);
            unsigned long long ga = (unsigned long long)(size_t)gsrc;
            unsigned lds_base = (unsigned)(size_t)(void*)in_tile;

            u32x4 g0;
            g0.x = 1u;                                   // count=1 valid user descriptor
            g0.y = lds_base;                             // lds_addr
            g0.z = (unsigned)(ga & 0xFFFFFFFFu);         // global_addr[31:0]
            g0.w = (unsigned)((ga >> 32) & 0x1FFFFFFu)   // global_addr[56:32]
                 | (2u << 30);                           // type = 2 (image/tensor)
            i32x8 g1;
            g1[0] = (2 << 16);                           // wg_mask=0, data_size=4B
            g1[1] = (WDIM & 0xFFFF) << 16;               // tensor_dim0 lo16 = 512
            g1[2] = (HDIM & 0xFFFF) << 16;               // dim0 hi=0 | tensor_dim1 lo16 = 512
            g1[3] = ICOLS << 16;                         // dim1 hi=0 | tile_dim0 = 82
            g1[4] = IROWS | (CCH << 16);                 // tile_dim1 = 12 | tile_dim2 = 16
            g1[5] = WDIM;                                // tensor_dim0_stride = 512
            g1[6] = 0;                                   // stride0 hi | stride1 lo16 (262144 lo16 = 0)
            g1[7] = (int)(plane >> 16);                  // tensor_dim1_stride[47:16] = 4
            i32x4 g2 = {CCH, 0, 0, 0};                   // tensor_dim2 = 16
            i32x4 g3 = {0, 0, 0, 0};
            i32x8 g4 = {0, 0, 0, 0, 0, 0, 0, 0};         // unused extra group (clang-23 6-arg form)
            __builtin_amdgcn_tensor_load_to_lds(g0, g1, g2, g3, g4, 0);
            __builtin_amdgcn_s_wait_tensorcnt(0);
        }
    } else
#endif
    {
        // Cooperative clamped staging (image-boundary tiles, or no-TDM fallback)
        for (int idx = tid; idx < CCH * IROWS * ICOLS; idx += 256) {
            const int c   = idx / (IROWS * ICOLS);
            const int rem = idx - c * (IROWS * ICOLS);
            const int r   = rem / ICOLS;
            const int ic  = rem - r * ICOLS;
            const int gh  = h0 - 2 + r;
            const int gw  = w0 - 9 + ic;
            const bool ok = (gh >= 0) && (gh < HDIM) && (gw >= 0) && (gw < WDIM);
            in_tile[idx] = ok ? xb[(size_t)c * plane + (size_t)gh * WDIM + gw] : 0.0f;
        }
    }

    // ---- Preload conv1 A-fragments (weights) into registers, ISA 16x4 f32 A layout ----
    // A[M=o][K=i] = w[o][i][kh][kw]; lane: M = lane%16, v0:K=k0+2*hi, v1:K=k0+2*hi+1
    v2f a1f[9][4];
    #pragma unroll
    for (int t = 0; t < 9; ++t) {
        #pragma unroll
        for (int c = 0; c < 4; ++c) {
            const int i0 = c * 4 + hi * 2;
            a1f[t][c].x = w1[(n * 16 + i0    ) * 9 + t];
            a1f[t][c].y = w1[(n * 16 + i0 + 1) * 9 + t];
        }
    }
    // Bias in C/D layout: VGPR j holds row o = j + 8*hi
    float bias1[8];
    #pragma unroll
    for (int j = 0; j < 8; ++j) bias1[j] = b1[j + hi * 8];

    __syncthreads();

    // ================= Stage 1: conv1 + bias + ReLU on haloed tile (from LDS) =================
    // 50 units of (mid-row m in [0,10), pixel-group g in [0,5)) over 8 waves.
    // Two interleaved accumulator chains break the serial WMMA RAW dependency.
    for (int u = wave; u < MROWS * 5; u += 8) {
        const int m = u / 5;
        const int g = u - m * 5;

        v8f acc0, acc1;
        #pragma unroll
        for (int j = 0; j < 8; ++j) { acc0[j] = bias1[j]; acc1[j] = 0.0f; }

        #pragma unroll
        for (int kh = 0; kh < 3; ++kh) {
            const int ir = m + kh;                       // in_tile row, 0..11
            #pragma unroll
            for (int kw = 0; kw < 3; ++kw) {
                const int ic = g * 16 + n + kw;          // in_tile col, 0..81
                #pragma unroll
                for (int c = 0; c < 4; ++c) {
                    const int i0 = c * 4 + hi * 2;       // B row K = i0 (v0) / i0+1 (v1)
                    v2f bf;
                    bf.x = in_tile[((i0    ) * IROWS + ir) * ICOLS + ic];
                    bf.y = in_tile[((i0 + 1) * IROWS + ir) * ICOLS + ic];
                    if (((kh * 3 + kw) * 4 + c) & 1)
                        acc1 = __builtin_amdgcn_wmma_f32_16x16x4_f32(
                            false, a1f[kh * 3 + kw][c], false, bf, (short)0, acc1, false, false);
                    else
                        acc0 = __builtin_amdgcn_wmma_f32_16x16x4_f32(
                            false, a1f[kh * 3 + kw][c], false, bf, (short)0, acc0, false, false);
                }
            }
        }

        // ReLU; zero out-of-image positions (conv2 then sees zero 'same' padding);
        // store channel-pairs as single b64 writes.
        const int  gh  = h0 - 1 + m;
        const int  gw  = w0 - 8 + g * 16 + n;
        const bool pok = (gh >= 0) && (gh < HDIM) && (gw >= 0) && (gw < WDIM);
        #pragma unroll
        for (int k = 0; k < 4; ++k) {
            float v0 = acc0[2 * k]     + acc1[2 * k];
            float v1 = acc0[2 * k + 1] + acc1[2 * k + 1];
            v0 = (v0 > 0.0f) ? v0 : 0.0f;
            v1 = (v1 > 0.0f) ? v1 : 0.0f;
            if (!pok) { v0 = 0.0f; v1 = 0.0f; }
            v2f pv; pv.x = v0; pv.y = v1;
            const int p = k + hi * 4;                    // channel pair = o>>1
            mid2[(p * MROWS + m) * MCOLS + g * 16 + n] = pv;
        }
    }

    // ---- Preload conv2 A-fragments + bias (after conv1 weights die -> lower peak VGPRs) ----
    v2f a2f[9][4];
    #pragma unroll
    for (int t = 0; t < 9; ++t) {
        #pragma unroll
        for (int c = 0; c < 4; ++c) {
            const int i0 = c * 4 + hi * 2;
            a2f[t][c].x = w2[(n * 16 + i0    ) * 9 + t];
            a2f[t][c].y = w2[(n * 16 + i0 + 1) * 9 + t];
        }
    }
    float bias2[8];
    #pragma unroll
    for (int j = 0; j < 8; ++j) bias2[j] = b2[j + hi * 8];

    __syncthreads();

    // ================= Stage 2: conv2 (from LDS, b64 pair fragments) + residual =================
    // 32 units of (row r in [0,8), group g in [0,4)) over 8 waves.
    for (int u = wave; u < HT * 4; u += 8) {
        const int r = u >> 2;
        const int g = u & 3;

        v8f acc0, acc1;
        #pragma unroll
        for (int j = 0; j < 8; ++j) { acc0[j] = bias2[j]; acc1[j] = 0.0f; }

        #pragma unroll
        for (int kh = 0; kh < 3; ++kh) {
            const int mm = r + kh;                       // mid row, 0..9
            #pragma unroll
            for (int kw = 0; kw < 3; ++kw) {
                const int gcol = g * 16 + n + kw + 7;    // mid col, 7..72
                #pragma unroll
                for (int c = 0; c < 4; ++c) {
                    const int p = c * 2 + hi;            // pair index -> rows (2p, 2p+1)
                    v2f bf = mid2[(p * MROWS + mm) * MCOLS + gcol];  // ds_load_b64
                    if (((kh * 3 + kw) * 4 + c) & 1)
                        acc1 = __builtin_amdgcn_wmma_f32_16x16x4_f32(
                            false, a2f[kh * 3 + kw][c], false, bf, (short)0, acc1, false, false);
                    else
                        acc0 = __builtin_amdgcn_wmma_f32_16x16x4_f32(
                            false, a2f[kh * 3 + kw][c], false, bf, (short)0, acc0, false, false);
                }
            }
        }

        const int hh = h0 + r;
        const int ww = w0 + g * 16 + n;
        #pragma unroll
        for (int j = 0; j < 8; ++j) {
            const int o = j + hi * 8;
            const size_t idx = (size_t)o * plane + (size_t)hh * WDIM + ww;
            ob[idx] = acc0[j] + acc1[j] + xb[idx];
        }
    }
}

extern "C" void kernel_launch(void* const* d_in, const int* in_sizes, int n_in,
                              void* d_out, int out_size, void* d_ws, size_t ws_size,
                              hipStream_t stream) {
    const float* x  = (const float*)d_in[0];
    const float* w1 = (const float*)d_in[1];
    const float* b1 = (const float*)d_in[2];
    const float* w2 = (const float*)d_in[3];
    const float* b2 = (const float*)d_in[4];
    float* out = (float*)d_out;

    // 16 batches x (512/8) row tiles x (512/64) col tiles = 8192 blocks
    dim3 grid(16 * 64 * 8);
    resblock_wmma_f32<<<grid, 256, 0, stream>>>(x, w1, b1, w2, b2, out);
}